// SOM_63316407878167
// MI455X (gfx1250) — compile-verified
//
#include <hip/hip_runtime.h>
#include <hip/hip_bf16.h>

typedef __attribute__((ext_vector_type(2))) float v2f;
typedef __attribute__((ext_vector_type(8))) float v8f;

constexpr int kMN    = 1024;   // M*N grid cells
constexpr int kDim   = 128;    // feature dim
constexpr int kBatch = 2048;   // batch

// ---------------------------------------------------------------------------
// Kernel 0: w2[m] = sum_d W[m,d]^2   (1024 rows, float4 loads)
// ---------------------------------------------------------------------------
__global__ void som_w2_kernel(const float* __restrict__ W, float* __restrict__ w2) {
    int row = blockIdx.x * blockDim.x + threadIdx.x;
    if (row >= kMN) return;
    const float4* wr = (const float4*)(W + row * kDim);
    float s = 0.f;
#pragma unroll 8
    for (int i = 0; i < kDim / 4; ++i) {
        float4 v = wr[i];
        s += v.x * v.x + v.y * v.y + v.z * v.z + v.w * v.w;
    }
    w2[row] = s;
}

// ---------------------------------------------------------------------------
// Kernel 1: BMU search. One wave per 16 batch columns. Loops over 64 mn-tiles,
// each tile = 32 x (V_WMMA_F32_16X16X4_F32) accumulating dot(W_row, x_col).
// score = w2[row] - 2*dot; running argmin per column; halves merged by shfl.
// Writes bmu grid location (as float2) per batch element.
// ---------------------------------------------------------------------------
__global__ void som_bmu_kernel(const float* __restrict__ W,
                               const float* __restrict__ X,
                               const float* __restrict__ w2,
                               const int*   __restrict__ loc,
                               float2*      __restrict__ bmu_loc) {
    const int lane  = threadIdx.x;        // 0..31 (wave32)
    const int col   = lane & 15;          // column within 16-wide batch tile
    const int half  = lane >> 4;          // 0: K 0,1 / rows 0-7;  1: K 2,3 / rows 8-15
    const int batch = blockIdx.x * 16 + col;

    const float* Xrow = X + batch * kDim; // B[k][n] = X[n][k] -> contiguous in k

    float best = 3.402823466e38f;
    int   bidx = 0;

    for (int mt = 0; mt < kMN / 16; ++mt) {
        const float* Wrow = W + (mt * 16 + col) * kDim;  // A row = lane&15
        v8f acc = {};
#pragma unroll 4
        for (int k0 = 0; k0 < kDim; k0 += 4) {
            const int kb = k0 + half * 2;                 // even -> aligned float2
            v2f a = *(const v2f*)(Wrow + kb);             // A[row][kb], A[row][kb+1]
            v2f b = *(const v2f*)(Xrow + kb);             // B[kb][col], B[kb+1][col]
            acc = __builtin_amdgcn_wmma_f32_16x16x4_f32(
                false, a, false, b, (short)0, acc, false, false);
        }
#pragma unroll
        for (int v = 0; v < 8; ++v) {                     // D: row v+8*half, col lane&15
            const int row = mt * 16 + v + half * 8;
            const float s = w2[row] - 2.0f * acc[v];
            if (s < best) { best = s; bidx = row; }
        }
    }
    // merge the two lane-halves (rows split 0-7 / 8-15 per tile);
    // tie-break on smaller index == jnp.argmin first-occurrence
    const float obest = __shfl_xor(best, 16);
    const int   oidx  = __shfl_xor(bidx, 16);
    if (obest < best || (obest == best && oidx < bidx)) { best = obest; bidx = oidx; }
    if (half == 0) {
        bmu_loc[batch] = make_float2((float)loc[bidx * 2], (float)loc[bidx * 2 + 1]);
    }
}

// ---------------------------------------------------------------------------
// Kernel 2: fused update. One wave per (16 mn-rows x 16 feature-cols) output
// tile. K-loop over batch in steps of 4: lr_op A-fragment generated on the fly
// (grid dist + exp), B-fragment from X, accumulate with f32 WMMA. Row-sums of
// lr_op accumulated per lane, merged via shfl. out = w + (lr@X) - rowsum*w.
// ---------------------------------------------------------------------------
__global__ void som_update_kernel(const float*  __restrict__ W,
                                  const float*  __restrict__ X,
                                  const int*    __restrict__ loc,
                                  const float2* __restrict__ bmu_loc,
                                  const int*    __restrict__ itp,
                                  float*        __restrict__ out) {
    const int lane = threadIdx.x;
    const int col  = lane & 15;           // feature column within tile
    const int half = lane >> 4;
    const int mt   = blockIdx.x;          // 0..63 mn tile
    const int d0   = blockIdx.y * 16;     // 0..7 feature tile * 16

    const float lr_decay   = 1.0f - (float)itp[0] * (1.0f / 100.0f);  // it/NITER
    const float alpha_op   = 0.3f * lr_decay;                          // ALPHA
    const float sigma_op   = 16.0f * lr_decay;                         // SIGMA=32/2
    const float neg_inv_s2 = -1.0f / (sigma_op * sigma_op);

    const int   arow = mt * 16 + col;     // A-matrix row this lane supplies
    const float lx = (float)loc[arow * 2];
    const float ly = (float)loc[arow * 2 + 1];

    v8f   acc  = {};
    float rsum = 0.f;                     // partial rowsum(lr_op) for row arow

    for (int k0 = 0; k0 < kBatch; k0 += 4) {
        const int kb = k0 + half * 2;     // even -> float4-aligned bmu_loc pair
        const float4 bl = *(const float4*)(&bmu_loc[kb]);  // (x_k,y_k,x_k1,y_k1)
        const float dx0 = lx - bl.x, dy0 = ly - bl.y;
        const float dx1 = lx - bl.z, dy1 = ly - bl.w;
        v2f a;
        a.x = alpha_op * __expf((dx0 * dx0 + dy0 * dy0) * neg_inv_s2);
        a.y = alpha_op * __expf((dx1 * dx1 + dy1 * dy1) * neg_inv_s2);
        rsum += a.x + a.y;
        v2f b;
        b.x = X[kb * kDim + d0 + col];
        b.y = X[(kb + 1) * kDim + d0 + col];
        acc = __builtin_amdgcn_wmma_f32_16x16x4_f32(
            false, a, false, b, (short)0, acc, false, false);
    }
    // full rowsum for row (lane&15): halves cover k%4 in {0,1} / {2,3}
    rsum += __shfl_xor(rsum, 16);
#pragma unroll
    for (int v = 0; v < 8; ++v) {
        const int   row = mt * 16 + v + half * 8;
        const float rs  = __shfl(rsum, v + half * 8);  // held by that lane
        const float w   = W[row * kDim + d0 + col];
        out[row * kDim + d0 + col] = w + acc[v] - rs * w;
    }
}

// ---------------------------------------------------------------------------
extern "C" void kernel_launch(void* const* d_in, const int* in_sizes, int n_in,
                              void* d_out, int out_size, void* d_ws, size_t ws_size,
                              hipStream_t stream) {
    const float* x        = (const float*)d_in[0];   // [2048,128] f32
    const float* weights  = (const float*)d_in[1];   // [1024,128] f32
    const int*   locations= (const int*)  d_in[2];   // [1024,2]   i32
    const int*   it       = (const int*)  d_in[3];   // [1]        i32
    float*       out      = (float*)d_out;           // [1024,128] f32

    float*  w2      = (float*)d_ws;                        // 1024 f32 (4 KB)
    float2* bmu_loc = (float2*)((char*)d_ws + 4096);       // 2048 float2 (16 KB)

    som_w2_kernel<<<kMN / 256, 256, 0, stream>>>(weights, w2);
    som_bmu_kernel<<<kBatch / 16, 32, 0, stream>>>(weights, x, w2, locations, bmu_loc);
    dim3 g2(kMN / 16, kDim / 16);
    som_update_kernel<<<g2, 32, 0, stream>>>(weights, x, locations, bmu_loc, it, out);
}